// DoubleQValueNet_31490700214937
// MI455X (gfx1250) — compile-verified
//
#include <hip/hip_runtime.h>

// ---------------------------------------------------------------------------
// CDNA5 (gfx1250) implementation: fp32 WMMA (V_WMMA_F32_16X16X4_F32) for all
// GEMMs, LDS-staged A tiles, pre-packed B fragments, fp32 global atomics for
// graph scatter.  wave32 everywhere.
// ---------------------------------------------------------------------------

typedef float v2f __attribute__((ext_vector_type(2)));
typedef float v8f __attribute__((ext_vector_type(8)));

#define LRELU(x) ((x) > 0.0f ? (x) : 0.01f * (x))

__device__ __forceinline__ v8f wmma4(v2f a, v2f b, v8f c) {
  // D = A(16x4 f32) * B(4x16 f32) + C(16x16 f32)
  return __builtin_amdgcn_wmma_f32_16x16x4_f32(false, a, false, b, (short)0, c,
                                               false, false);
}

__device__ __forceinline__ void atomAddF(float* p, float v) {
  __hip_atomic_fetch_add(p, v, __ATOMIC_RELAXED, __HIP_MEMORY_SCOPE_AGENT);
}

// 16 rows x 64 cols tile, K = 64 (16 WMMA K-steps).  A in LDS (ldA stride),
// B pre-packed in lane order: pk[(step*4 + tile)*32 + lane] as float2.
__device__ __forceinline__ void gemm_k64(const float* __restrict__ A, int ldA,
                                         int arow, int kb,
                                         const v2f* __restrict__ B,
                                         v8f* acc) {
  int lane = threadIdx.x & 31;
#pragma unroll
  for (int s = 0; s < 16; ++s) {
    v2f a;
    a.x = A[arow * ldA + s * 4 + kb];
    a.y = A[arow * ldA + s * 4 + kb + 1];
#pragma unroll
    for (int t = 0; t < 4; ++t) {
      v2f b = B[(s * 4 + t) * 32 + lane];
      acc[t] = wmma4(a, b, acc[t]);
    }
  }
}

// ---------------------------------------------------------------------------
__global__ void zero_kernel(float* p, long n) {
  long i = (long)blockIdx.x * blockDim.x + threadIdx.x;
  long st = (long)gridDim.x * blockDim.x;
  for (; i < n; i += st) p[i] = 0.0f;
}

// Pack W (KxN row-major) into WMMA-B fragment order:
// pk[(step*(N/16)+tile)*32+lane] = {W[k][n], W[k+1][n]},
//   n = tile*16 + (lane&15), k = step*4 + ((lane>>4)<<1)
__global__ void pack_kernel(const float* __restrict__ W, int K, int N,
                            float* __restrict__ outp) {
  long total = (long)(K / 4) * (N / 16) * 32;
  long i = (long)blockIdx.x * blockDim.x + threadIdx.x;
  long st = (long)gridDim.x * blockDim.x;
  int ntiles = N / 16;
  for (; i < total; i += st) {
    int lane = (int)(i & 31);
    long q = i >> 5;
    int tile = (int)(q % ntiles);
    int step = (int)(q / ntiles);
    int n = tile * 16 + (lane & 15);
    int k = step * 4 + ((lane >> 4) << 1);
    v2f v;
    v.x = W[(long)k * N + n];
    v.y = W[(long)(k + 1) * N + n];
    ((v2f*)outp)[i] = v;
  }
}

// ---------------------------------------------------------------------------
// q_gcnn: t = lrelu( lrelu(x[src]@W1a + x[dst]@W1b + ang*w128 + act*w129
//                          + gt*w130 + b1) @ W2 + b2 )
__global__ __launch_bounds__(256) void qgcnn_kernel(
    const float* __restrict__ x, const int* __restrict__ srcI,
    const int* __restrict__ dstI, const float* __restrict__ ang,
    const float* __restrict__ act, const float* __restrict__ gt,
    const float* __restrict__ w1raw, const float* __restrict__ b1,
    const float* __restrict__ pkA, const float* __restrict__ pkB,
    const float* __restrict__ pkW2, const float* __restrict__ b2,
    float* __restrict__ out, long E) {
  __shared__ float xs[128 * 65];
  __shared__ float xd[128 * 65];
  __shared__ float hid[128 * 65];
  __shared__ float sang[128], sact[128], sgt[128];
  int tid = threadIdx.x;
  long base = (long)blockIdx.x * 128;
  for (int i = tid; i < 128 * 64; i += 256) {
    int r = i >> 6, c = i & 63;
    long e = base + r;
    if (e >= E) e = E - 1;
    xs[r * 65 + c] = x[(long)srcI[e] * 64 + c];
    xd[r * 65 + c] = x[(long)dstI[e] * 64 + c];
  }
  for (int i = tid; i < 128; i += 256) {
    long e = base + i;
    if (e >= E) e = E - 1;
    sang[i] = ang[e];
    sact[i] = act[e];
    sgt[i] = gt[e];
  }
  __syncthreads();
  int lane = tid & 31, wave = tid >> 5;
  int rowbase = wave * 16;
  int arow = rowbase + (lane & 15);
  int kb = (lane >> 4) << 1;
  int rhalf = (lane >> 4) << 3;
  v8f zz = {0, 0, 0, 0, 0, 0, 0, 0};
  v8f acc[4] = {zz, zz, zz, zz};
  gemm_k64(xs, 65, arow, kb, (const v2f*)pkA, acc);
  gemm_k64(xd, 65, arow, kb, (const v2f*)pkB, acc);
#pragma unroll
  for (int t = 0; t < 4; ++t) {
    int col = t * 16 + (lane & 15);
    float w128 = w1raw[128 * 64 + col];
    float w129 = w1raw[129 * 64 + col];
    float w130 = w1raw[130 * 64 + col];
    float bb = b1[col];
#pragma unroll
    for (int r = 0; r < 8; ++r) {
      int row = rowbase + r + rhalf;
      float v = acc[t][r] + sang[row] * w128 + sact[row] * w129 +
                sgt[row] * w130 + bb;
      hid[row * 65 + col] = LRELU(v);
    }
  }
  __syncthreads();
  v8f acc2[4] = {zz, zz, zz, zz};
  gemm_k64(hid, 65, arow, kb, (const v2f*)pkW2, acc2);
#pragma unroll
  for (int t = 0; t < 4; ++t) {
    int col = t * 16 + (lane & 15);
    float bb = b2[col];
#pragma unroll
    for (int r = 0; r < 8; ++r) {
      long e = base + rowbase + r + rhalf;
      if (e < E) out[e * 64 + col] = LRELU(acc2[t][r] + bb);
    }
  }
}

// ---------------------------------------------------------------------------
// batchnorm (biased variance, eps=1e-5)
__global__ __launch_bounds__(256) void bn_stats_kernel(
    const float* __restrict__ t, long rows, float* __restrict__ stats) {
  __shared__ float s1[256], s2[256];
  int tid = threadIdx.x;
  int c = tid & 63, rg = tid >> 6;
  long base = (long)blockIdx.x * 1024;
  float a = 0.f, b = 0.f;
  for (int r = rg; r < 1024; r += 4) {
    long row = base + r;
    if (row < rows) {
      float v = t[row * 64 + c];
      a += v;
      b += v * v;
    }
  }
  s1[tid] = a;
  s2[tid] = b;
  __syncthreads();
  if (tid < 64) {
    float A = s1[tid] + s1[tid + 64] + s1[tid + 128] + s1[tid + 192];
    float B = s2[tid] + s2[tid + 64] + s2[tid + 128] + s2[tid + 192];
    atomAddF(&stats[tid], A);
    atomAddF(&stats[64 + tid], B);
  }
}

__global__ void bn_finalize_kernel(const float* __restrict__ stats,
                                   const float* __restrict__ g,
                                   const float* __restrict__ b, float rows,
                                   float* __restrict__ scsh) {
  int c = threadIdx.x;
  if (c < 64) {
    float m = stats[c] / rows;
    float v = stats[64 + c] / rows - m * m;
    float sc = g[c] * rsqrtf(v + 1e-5f);
    scsh[c] = sc;
    scsh[64 + c] = b[c] - m * sc;
  }
}

__global__ void bn_apply_kernel(float* __restrict__ t,
                                const float* __restrict__ scsh, long n) {
  long i = (long)blockIdx.x * blockDim.x + threadIdx.x;
  if (i < n) {
    int c = (int)(i & 63);
    t[i] = t[i] * scsh[c] + scsh[64 + c];
  }
}

// ---------------------------------------------------------------------------
__global__ void deg_edge_kernel(const int* __restrict__ s,
                                const int* __restrict__ d,
                                float* __restrict__ deg, long E) {
  long i = (long)blockIdx.x * blockDim.x + threadIdx.x;
  if (i < E) {
    atomAddF(&deg[s[i]], 1.0f);
    atomAddF(&deg[d[i]], 1.0f);
  }
}

__global__ void deg_pair_kernel(const int* __restrict__ sep,
                                float* __restrict__ deg, long G) {
  long i = (long)blockIdx.x * blockDim.x + threadIdx.x;
  if (i < G) {
    atomAddF(&deg[sep[2 * i]], 1.0f);
    atomAddF(&deg[sep[2 * i + 1]], 1.0f);
  }
}

__global__ void recip_kernel(float* __restrict__ deg, long n) {
  long i = (long)blockIdx.x * blockDim.x + threadIdx.x;
  if (i < n) deg[i] = 1.0f / fmaxf(deg[i], 1.0f);
}

// node[src[e]] += h[e]; node[dst[e]] += h[e]   (edge-feature scatter)
__global__ void scatter_edge_kernel(const float* __restrict__ h,
                                    const int* __restrict__ s,
                                    const int* __restrict__ d,
                                    float* __restrict__ node, long E) {
  long i = (long)blockIdx.x * blockDim.x + threadIdx.x;
  long tot = E * 16;
  if (i >= tot) return;
  long e = i >> 4;
  int cg = (int)(i & 15) * 4;
  float4 v = *(const float4*)(h + e * 64 + cg);
  float* n1 = node + (long)s[e] * 64 + cg;
  float* n2 = node + (long)d[e] * 64 + cg;
  atomAddF(n1 + 0, v.x); atomAddF(n1 + 1, v.y);
  atomAddF(n1 + 2, v.z); atomAddF(n1 + 3, v.w);
  atomAddF(n2 + 0, v.x); atomAddF(n2 + 1, v.y);
  atomAddF(n2 + 2, v.z); atomAddF(n2 + 3, v.w);
}

// symmetric pair scatter: node[a] += h[b]; node[b] += h[a]
__global__ void scatter_pair_kernel(const float* __restrict__ h,
                                    const int* __restrict__ sep,
                                    float* __restrict__ node, long G) {
  long i = (long)blockIdx.x * blockDim.x + threadIdx.x;
  long tot = G * 16;
  if (i >= tot) return;
  long j = i >> 4;
  int cg = (int)(i & 15) * 4;
  int a = sep[2 * j], b = sep[2 * j + 1];
  float4 va = *(const float4*)(h + (long)a * 64 + cg);
  float4 vb = *(const float4*)(h + (long)b * 64 + cg);
  float* na = node + (long)a * 64 + cg;
  float* nb = node + (long)b * 64 + cg;
  atomAddF(na + 0, vb.x); atomAddF(na + 1, vb.y);
  atomAddF(na + 2, vb.z); atomAddF(na + 3, vb.w);
  atomAddF(nb + 0, va.x); atomAddF(nb + 1, va.y);
  atomAddF(nb + 2, va.z); atomAddF(nb + 3, va.w);
}

// GCN layer GEMM: z -> lrelu(z @ W + b).
// gatherMode=1: z[e] = node[src[e]]*inv[src] + node[dst[e]]*inv[dst]
// gatherMode=0: z[v] = node[v]*inv[v]
__global__ __launch_bounds__(256) void gcn_gemm_kernel(
    const float* __restrict__ node, const float* __restrict__ invdeg,
    const int* __restrict__ sA, const int* __restrict__ sB,
    const float* __restrict__ pkW, const float* __restrict__ bias,
    float* __restrict__ out, long rows, int gatherMode) {
  __shared__ float z[128 * 65];
  int tid = threadIdx.x;
  long base = (long)blockIdx.x * 128;
  for (int i = tid; i < 128 * 64; i += 256) {
    int r = i >> 6, c = i & 63;
    long row = base + r;
    if (row >= rows) row = rows - 1;
    float v;
    if (gatherMode) {
      int a = sA[row], b = sB[row];
      v = node[(long)a * 64 + c] * invdeg[a] +
          node[(long)b * 64 + c] * invdeg[b];
    } else {
      v = node[row * 64 + c] * invdeg[row];
    }
    z[r * 65 + c] = v;
  }
  __syncthreads();
  int lane = tid & 31, wave = tid >> 5;
  int rowbase = wave * 16;
  int arow = rowbase + (lane & 15);
  int kb = (lane >> 4) << 1;
  int rhalf = (lane >> 4) << 3;
  v8f zz = {0, 0, 0, 0, 0, 0, 0, 0};
  v8f acc[4] = {zz, zz, zz, zz};
  gemm_k64(z, 65, arow, kb, (const v2f*)pkW, acc);
#pragma unroll
  for (int t = 0; t < 4; ++t) {
    int col = t * 16 + (lane & 15);
    float bb = bias[col];
#pragma unroll
    for (int r = 0; r < 8; ++r) {
      long g = base + rowbase + r + rhalf;
      if (g < rows) out[g * 64 + col] = LRELU(acc[t][r] + bb);
    }
  }
}

// ---------------------------------------------------------------------------
__global__ void gather_kernel(const float* __restrict__ sg,
                              const int* __restrict__ idx,
                              float* __restrict__ out, long n) {
  long i = (long)blockIdx.x * blockDim.x + threadIdx.x;
  if (i < n) {
    long g = i >> 6;
    int c = (int)(i & 63);
    out[i] = sg[(long)idx[g] * 64 + c];
  }
}

__global__ void invcnt_kernel(const int* __restrict__ eoffs,
                              float* __restrict__ invcnt, int B) {
  int i = threadIdx.x;
  if (i < B) invcnt[i] = 1.0f / (float)(eoffs[i + 1] - eoffs[i]);
}

__global__ void segmean_kernel(const float* __restrict__ gg,
                               const int* __restrict__ eoffs,
                               const float* __restrict__ invcnt,
                               float* __restrict__ out, long E, int B) {
  long i = (long)blockIdx.x * blockDim.x + threadIdx.x;
  long tot = E * 64;
  if (i >= tot) return;
  long e = i >> 6;
  int c = (int)(i & 63);
  int seg = 0;
  for (int s = 1; s <= B; ++s)
    if (e >= (long)eoffs[s]) seg = s;
  if (seg > B - 1) seg = B - 1;
  atomAddF(&out[seg * 64 + c], gg[i] * invcnt[seg]);
}

// extra lrelu fused + mean over S=16 group
__global__ void pmean_kernel(const float* __restrict__ h,
                             float* __restrict__ p, long nsub) {
  long i = (long)blockIdx.x * blockDim.x + threadIdx.x;
  if (i >= nsub * 64) return;
  long s = i >> 6;
  int c = (int)(i & 63);
  float acc = 0.f;
  for (int r = 0; r < 16; ++r) {
    float v = h[(s * 16 + r) * 64 + c];
    acc += LRELU(v);
  }
  p[i] = acc * (1.0f / 16.0f);
}

// tiny dense (8-row value MLPs): scalar VALU, negligible work
__global__ void dense_kernel(const float* __restrict__ X,
                             const float* __restrict__ W,
                             const float* __restrict__ b,
                             float* __restrict__ out, int rows, int K, int N,
                             int act) {
  long i = (long)blockIdx.x * blockDim.x + threadIdx.x;
  long tot = (long)rows * N;
  if (i >= tot) return;
  int r = (int)(i / N), n = (int)(i % N);
  float s = b[n];
  for (int k = 0; k < K; ++k) s += X[(long)r * K + k] * W[(long)k * N + n];
  out[i] = act ? LRELU(s) : s;
}

// ---------------------------------------------------------------------------
// Fused 64 -> 1024 -> 1024 -> 64 MLP, one 16-row tile per block.
// LDS: p(16x65) + h1(16x1040) + h2(16x1040) = ~137 KB of the 320 KB WGP LDS.
__global__ __launch_bounds__(256) void mlp_big_kernel(
    const float* __restrict__ P, const float* __restrict__ pkW1,
    const float* __restrict__ b1, const float* __restrict__ pkW2,
    const float* __restrict__ b2, const float* __restrict__ pkW3,
    const float* __restrict__ b3, float* __restrict__ out, long rows) {
  __shared__ float pt[16 * 65];
  __shared__ float h1[16 * 1040];
  __shared__ float h2[16 * 1040];
  int tid = threadIdx.x;
  long base = (long)blockIdx.x * 16;
  for (int i = tid; i < 16 * 64; i += 256) {
    int r = i >> 6, c = i & 63;
    long row = base + r;
    if (row >= rows) row = rows - 1;
    pt[r * 65 + c] = P[row * 64 + c];
  }
  __syncthreads();
  int lane = tid & 31, wave = tid >> 5;
  int arow = lane & 15;
  int kb = (lane >> 4) << 1;
  int rhalf = (lane >> 4) << 3;
  const v2f* W1 = (const v2f*)pkW1;
  const v2f* W2 = (const v2f*)pkW2;
  const v2f* W3 = (const v2f*)pkW3;
  v8f zz = {0, 0, 0, 0, 0, 0, 0, 0};
  // L1: K=64, N=1024 (64 tiles, 8 per wave)
#pragma unroll
  for (int tt = 0; tt < 8; ++tt) {
    int tile = wave * 8 + tt;
    v8f acc = zz;
#pragma unroll
    for (int s = 0; s < 16; ++s) {
      v2f a;
      a.x = pt[arow * 65 + s * 4 + kb];
      a.y = pt[arow * 65 + s * 4 + kb + 1];
      v2f b = W1[(s * 64 + tile) * 32 + lane];
      acc = wmma4(a, b, acc);
    }
    int col = tile * 16 + (lane & 15);
    float bb = b1[col];
#pragma unroll
    for (int r = 0; r < 8; ++r) h1[(r + rhalf) * 1040 + col] = LRELU(acc[r] + bb);
  }
  __syncthreads();
  // L2: K=1024, N=1024
  for (int tt = 0; tt < 8; ++tt) {
    int tile = wave * 8 + tt;
    v8f acc = zz;
    for (int s = 0; s < 256; ++s) {
      v2f a;
      a.x = h1[arow * 1040 + s * 4 + kb];
      a.y = h1[arow * 1040 + s * 4 + kb + 1];
      v2f b = W2[((long)s * 64 + tile) * 32 + lane];
      acc = wmma4(a, b, acc);
    }
    int col = tile * 16 + (lane & 15);
    float bb = b2[col];
#pragma unroll
    for (int r = 0; r < 8; ++r) h2[(r + rhalf) * 1040 + col] = LRELU(acc[r] + bb);
  }
  __syncthreads();
  // L3: K=1024, N=64 (4 tiles, waves 0..3)
  if (wave < 4) {
    int tile = wave;
    v8f acc = zz;
    for (int s = 0; s < 256; ++s) {
      v2f a;
      a.x = h2[arow * 1040 + s * 4 + kb];
      a.y = h2[arow * 1040 + s * 4 + kb + 1];
      v2f b = W3[(s * 4 + tile) * 32 + lane];
      acc = wmma4(a, b, acc);
    }
    int col = tile * 16 + (lane & 15);
    float bb = b3[col];
#pragma unroll
    for (int r = 0; r < 8; ++r) {
      long row = base + r + rhalf;
      if (row < rows) out[row * 64 + col] = acc[r] + bb;
    }
  }
}

// ---------------------------------------------------------------------------
extern "C" void kernel_launch(void* const* d_in, const int* in_sizes, int n_in,
                              void* d_out, int out_size, void* d_ws,
                              size_t ws_size, hipStream_t stream) {
  (void)n_in; (void)out_size; (void)ws_size;
  auto Fp = [&](int i) { return (const float*)d_in[i]; };
  auto Ip = [&](int i) { return (const int*)d_in[i]; };

  const float* x = Fp(0);
  const float* actions = Fp(1);
  const int* eidx = Ip(2);
  const float* angles = Fp(3);
  const int* subg = Ip(4);
  const int* sep = Ip(5);
  const int* eoffs = Ip(6);
  const float* gte = Fp(7);
  const float *q1w1 = Fp(9), *q1b1 = Fp(10), *q1w2 = Fp(11), *q1b2 = Fp(12);
  const float *q2w1 = Fp(13), *q2b1 = Fp(14), *q2w2 = Fp(15), *q2b2 = Fp(16);
  const float *bng = Fp(17), *bnb = Fp(18);
  const float *g13W = Fp(19), *g13b = Fp(20), *g23W = Fp(21), *g23b = Fp(22);
  const float *g12W = Fp(23), *g12b = Fp(24), *g22W = Fp(25), *g22b = Fp(26);
  const float* vw[4][6];
  for (int v = 0; v < 4; ++v)
    for (int j = 0; j < 6; ++j) vw[v][j] = Fp(27 + v * 6 + j);
  // vw[0]=v1, vw[1]=v12, vw[2]=v2, vw[3]=v22

  const long E = in_sizes[1];
  const long N = in_sizes[0] / 64;
  const long G = in_sizes[4];
  const int Bseg = in_sizes[6] - 1;
  const long NSUB = G / 16;
  const int DG = (int)(in_sizes[19] / 4096);  // 20
  const int DS = (int)(in_sizes[23] / 4096);  // 10

  const int* srcI = eidx;
  const int* dstI = eidx + E;

  float* ws = (float*)d_ws;
  size_t off = 0;
  auto alloc = [&](size_t n) {
    float* p = ws + off;
    off += (n + 255) & ~(size_t)255;
    return p;
  };
  float* sg1 = alloc((size_t)E * 64);
  float* sg2 = alloc((size_t)E * 64);
  float* bufA = alloc((size_t)G * 64);
  float* bufB = alloc((size_t)G * 64);
  float* node = alloc((size_t)G * 64);
  float* deg = alloc((size_t)G);
  float* stats = alloc(128);
  float* scsh = alloc(128);
  float* invc = alloc(64);
  float* ggm1 = alloc(512);
  float* ggm2 = alloc(512);
  float* h1s = alloc(8 * 1024);
  float* h2s = alloc(8 * 1024);
  float* pbuf = alloc((size_t)NSUB * 64);
  float* pkq1a = alloc(4096); float* pkq1b = alloc(4096); float* pkq1w2 = alloc(4096);
  float* pkq2a = alloc(4096); float* pkq2b = alloc(4096); float* pkq2w2 = alloc(4096);
  float* pkg13 = alloc((size_t)DG * 4096);
  float* pkg23 = alloc((size_t)DG * 4096);
  float* pkg12 = alloc((size_t)DS * 4096);
  float* pkg22 = alloc((size_t)DS * 4096);
  float* pkv1w1 = alloc(64 * 1024);
  float* pkv1w2 = alloc((size_t)1024 * 1024);
  float* pkv1w3 = alloc(1024 * 64);
  float* pkv2w1 = alloc(64 * 1024);
  float* pkv2w2 = alloc((size_t)1024 * 1024);
  float* pkv2w3 = alloc(1024 * 64);

  float* outp = (float*)d_out;
  const size_t o_q1 = 0;
  const size_t o_v12 = (size_t)NSUB * 64;
  const size_t o_q2 = o_v12 + (size_t)Bseg * 64;
  const size_t o_v22 = o_q2 + (size_t)NSUB * 64;

  auto cd = [](long n, long per) { return (unsigned)((n + per - 1) / per); };

  // ---- pack weights into WMMA B-fragment order (all L2-resident) ----------
  auto pack = [&](const float* W, int K, int Nc, float* out_) {
    pack_kernel<<<cd((long)K * Nc / 2, 256), 256, 0, stream>>>(W, K, Nc, out_);
  };
  pack(q1w1, 64, 64, pkq1a);
  pack(q1w1 + 64 * 64, 64, 64, pkq1b);
  pack(q1w2, 64, 64, pkq1w2);
  pack(q2w1, 64, 64, pkq2a);
  pack(q2w1 + 64 * 64, 64, 64, pkq2b);
  pack(q2w2, 64, 64, pkq2w2);
  for (int l = 0; l < DG; ++l) {
    pack(g13W + (size_t)l * 4096, 64, 64, pkg13 + (size_t)l * 4096);
    pack(g23W + (size_t)l * 4096, 64, 64, pkg23 + (size_t)l * 4096);
  }
  for (int l = 0; l < DS; ++l) {
    pack(g12W + (size_t)l * 4096, 64, 64, pkg12 + (size_t)l * 4096);
    pack(g22W + (size_t)l * 4096, 64, 64, pkg22 + (size_t)l * 4096);
  }
  pack(vw[0][0], 64, 1024, pkv1w1);
  pack(vw[0][2], 1024, 1024, pkv1w2);
  pack(vw[0][4], 1024, 64, pkv1w3);
  pack(vw[2][0], 64, 1024, pkv2w1);
  pack(vw[2][2], 1024, 1024, pkv2w2);
  pack(vw[2][4], 1024, 64, pkv2w3);

  // ---- Phase A: edge MLPs (q_gcnn) + batchnorm ----------------------------
  qgcnn_kernel<<<cd(E, 128), 256, 0, stream>>>(x, srcI, dstI, angles, actions,
                                               gte, q1w1, q1b1, pkq1a, pkq1b,
                                               pkq1w2, q1b2, sg1, E);
  qgcnn_kernel<<<cd(E, 128), 256, 0, stream>>>(x, srcI, dstI, angles, actions,
                                               gte, q2w1, q2b1, pkq2a, pkq2b,
                                               pkq2w2, q2b2, sg2, E);
  for (int q = 0; q < 2; ++q) {
    float* sg = q ? sg2 : sg1;
    zero_kernel<<<1, 256, 0, stream>>>(stats, 128);
    bn_stats_kernel<<<cd(E, 1024), 256, 0, stream>>>(sg, E, stats);
    bn_finalize_kernel<<<1, 64, 0, stream>>>(stats, bng, bnb, (float)E, scsh);
    bn_apply_kernel<<<cd(E * 64, 256), 256, 0, stream>>>(sg, scsh, E * 64);
  }

  // ---- Phase B: two 20-layer GCNs over (N nodes, E edges), both on sg1 ----
  zero_kernel<<<2048, 256, 0, stream>>>(deg, N);
  deg_edge_kernel<<<cd(E, 256), 256, 0, stream>>>(srcI, dstI, deg, E);
  recip_kernel<<<cd(N, 256), 256, 0, stream>>>(deg, N);
  invcnt_kernel<<<1, 64, 0, stream>>>(eoffs, invc, Bseg);
  for (int net = 0; net < 2; ++net) {
    const float* pk = net ? pkg23 : pkg13;
    const float* gb = net ? g23b : g13b;
    float* ggm = net ? ggm2 : ggm1;
    const float* hcur = sg1;  // both gg nets consume sg1_det
    for (int l = 0; l < DG; ++l) {
      zero_kernel<<<2048, 256, 0, stream>>>(node, N * 64);
      scatter_edge_kernel<<<cd(E * 16, 256), 256, 0, stream>>>(hcur, srcI,
                                                               dstI, node, E);
      float* hout = (l & 1) ? bufB : bufA;
      gcn_gemm_kernel<<<cd(E, 128), 256, 0, stream>>>(
          node, deg, srcI, dstI, pk + (size_t)l * 4096, gb + (size_t)l * 64,
          hout, E, 1);
      hcur = hout;
    }
    zero_kernel<<<1, 256, 0, stream>>>(ggm, 512);
    segmean_kernel<<<cd(E * 64, 256), 256, 0, stream>>>(hcur, eoffs, invc, ggm,
                                                        E, Bseg);
  }
  // tiny value MLPs (8 rows)
  {
    const float** w = vw[1];  // v12
    dense_kernel<<<cd((long)Bseg * 1024, 256), 256, 0, stream>>>(
        ggm1, w[0], w[1], h1s, Bseg, 64, 1024, 1);
    dense_kernel<<<cd((long)Bseg * 1024, 256), 256, 0, stream>>>(
        h1s, w[2], w[3], h2s, Bseg, 1024, 1024, 1);
    dense_kernel<<<cd((long)Bseg * 64, 256), 256, 0, stream>>>(
        h2s, w[4], w[5], outp + o_v12, Bseg, 1024, 64, 0);
  }
  {
    const float** w = vw[3];  // v22
    dense_kernel<<<cd((long)Bseg * 1024, 256), 256, 0, stream>>>(
        ggm2, w[0], w[1], h1s, Bseg, 64, 1024, 1);
    dense_kernel<<<cd((long)Bseg * 1024, 256), 256, 0, stream>>>(
        h1s, w[2], w[3], h2s, Bseg, 1024, 1024, 1);
    dense_kernel<<<cd((long)Bseg * 64, 256), 256, 0, stream>>>(
        h2s, w[4], w[5], outp + o_v22, Bseg, 1024, 64, 0);
  }

  // ---- Phase C: subgraph GCNs (GATHER nodes) + big MLPs -------------------
  zero_kernel<<<2048, 256, 0, stream>>>(deg, G);
  deg_pair_kernel<<<cd(G, 256), 256, 0, stream>>>(sep, deg, G);
  recip_kernel<<<cd(G, 256), 256, 0, stream>>>(deg, G);
  for (int net = 0; net < 2; ++net) {
    const float* sg = net ? sg2 : sg1;
    const float* pk = net ? pkg22 : pkg12;
    const float* gb = net ? g22b : g12b;
    const float* pw1 = net ? pkv2w1 : pkv1w1;
    const float* pw2 = net ? pkv2w2 : pkv1w2;
    const float* pw3 = net ? pkv2w3 : pkv1w3;
    const float** w = vw[net ? 2 : 0];  // v2 / v1 biases
    float* qout = outp + (net ? o_q2 : o_q1);

    gather_kernel<<<cd(G * 64, 256), 256, 0, stream>>>(sg, subg, bufA, G * 64);
    float* hcur = bufA;
    for (int l = 0; l < DS; ++l) {
      zero_kernel<<<2048, 256, 0, stream>>>(node, G * 64);
      scatter_pair_kernel<<<cd(G * 16, 256), 256, 0, stream>>>(hcur, sep, node,
                                                               G);
      float* hout = (hcur == bufA) ? bufB : bufA;
      gcn_gemm_kernel<<<cd(G, 128), 256, 0, stream>>>(
          node, deg, srcI, dstI, pk + (size_t)l * 4096, gb + (size_t)l * 64,
          hout, G, 0);
      hcur = hout;
    }
    pmean_kernel<<<cd(NSUB * 64, 256), 256, 0, stream>>>(hcur, pbuf, NSUB);
    mlp_big_kernel<<<cd(NSUB, 16), 256, 0, stream>>>(
        pbuf, pw1, w[1], pw2, w[3], pw3, w[5], qout, NSUB);
  }
}